// attention_10144712753519
// MI455X (gfx1250) — compile-verified
//
#include <hip/hip_runtime.h>
#include <math.h>

#define NN 50000
#define NE 800000
#define DIMV 64
#define NH 2
#define HDV 128

typedef __attribute__((ext_vector_type(2))) float v2f;
typedef __attribute__((ext_vector_type(8))) float v8f;

// ---------------- workspace layout (float offsets) ----------------
#define OFF_Q    ((size_t)0)
#define OFF_K    (OFF_Q    + (size_t)NN * HDV)     //  6,400,000
#define OFF_V    (OFF_K    + (size_t)NN * HDV)
#define OFF_SKIP (OFF_V    + (size_t)NN * HDV)
#define OFF_AGG  (OFF_SKIP + (size_t)NN * HDV)
#define OFF_ATTN (OFF_AGG  + (size_t)NN * HDV)     // E*H
#define OFF_AMAX (OFF_ATTN + (size_t)NE * NH)      // N*H
#define OFF_ASUM (OFF_AMAX + (size_t)NN * NH)
#define OFF_Y    (OFF_ASUM + (size_t)NN * NH)      // N*DIM
#define OFF_SUMS (OFF_Y    + (size_t)NN * DIMV)    // 2 scalars

// order-preserving float atomic max via int/uint punning (init = -inf)
__device__ inline void atomicMaxFloat(float* addr, float val) {
    if (val >= 0.0f) {
        atomicMax((int*)addr, __float_as_int(val));
    } else {
        atomicMin((unsigned int*)addr, __float_as_uint(val));
    }
}

// ---------------- init accumulators ----------------
__global__ __launch_bounds__(256)
void init_kernel(float* __restrict__ agg, float* __restrict__ amax,
                 float* __restrict__ asum, float* __restrict__ sums) {
    const int i = blockIdx.x * blockDim.x + threadIdx.x;
    if (i < NN * HDV) agg[i] = 0.0f;
    if (i < NN * NH) { amax[i] = -__builtin_inff(); asum[i] = 0.0f; }
    if (i < 2) sums[i] = 0.0f;
}

// ---------------- fused q/k/v/skip projection via WMMA f32 16x16x4 ----------------
// grid (3125, 8), block 128 (4 waves). wave -> one 16x16 output tile of one projection.
__global__ __launch_bounds__(128)
void proj_kernel(const float* __restrict__ x,
                 const float* __restrict__ Wq, const float* __restrict__ bq,
                 const float* __restrict__ Wk, const float* __restrict__ bk,
                 const float* __restrict__ Wv, const float* __restrict__ bv,
                 const float* __restrict__ Ws, const float* __restrict__ bs,
                 float* __restrict__ q, float* __restrict__ k,
                 float* __restrict__ v, float* __restrict__ s) {
    const int lane  = threadIdx.x & 31;
    const int wave  = threadIdx.x >> 5;
    const int tile  = blockIdx.y * 4 + wave;   // 0..31
    const int proj  = tile >> 3;               // 0..3 : q,k,v,skip
    const int cb    = (tile & 7) * 16;         // column base within HD
    const int rb    = blockIdx.x * 16;         // row base (N = 3125*16 exact)
    const int m     = lane & 15;
    const int kh    = lane >> 4;

    const float* W; const float* B; float* O;
    switch (proj) {
        case 0:  W = Wq; B = bq; O = q; break;
        case 1:  W = Wk; B = bk; O = k; break;
        case 2:  W = Wv; B = bv; O = v; break;
        default: W = Ws; B = bs; O = s; break;
    }
    const int col = cb + m;
    const float* xrow = x + (size_t)(rb + m) * DIMV;

    v8f acc = {};
#pragma unroll
    for (int kk = 0; kk < 16; ++kk) {
        const int kb = kk * 4 + kh * 2;
        v2f a, b;
        a[0] = xrow[kb];
        a[1] = xrow[kb + 1];
        b[0] = W[(size_t)kb * HDV + col];
        b[1] = W[(size_t)(kb + 1) * HDV + col];
        acc = __builtin_amdgcn_wmma_f32_16x16x4_f32(false, a, false, b,
                                                    (short)0, acc, false, false);
    }
    const float bias = B[col];
#pragma unroll
    for (int r = 0; r < 8; ++r) {
        const int row = rb + r + kh * 8;
        O[(size_t)row * HDV + col] = acc[r] + bias;
    }
}

// ---------------- edge logits + segment max ----------------
__global__ __launch_bounds__(256)
void edge_logits_kernel(const int* __restrict__ ei,
                        const float* __restrict__ q, const float* __restrict__ k,
                        float* __restrict__ attn, float* __restrict__ amax) {
    const int t = blockIdx.x * blockDim.x + threadIdx.x;
    if (t >= NE * NH) return;
    const int e = t >> 1;
    const int h = t & 1;
    const int src = ei[e];
    const int dst = ei[NE + e];
    const float* qd = q + (size_t)dst * HDV + h * DIMV;
    const float* ks = k + (size_t)src * HDV + h * DIMV;
    float acc = 0.0f;
#pragma unroll 4
    for (int d = 0; d < DIMV; d += 4) {
        const float4 qa = *(const float4*)(qd + d);
        const float4 ka = *(const float4*)(ks + d);
        acc += qa.x * ka.x + qa.y * ka.y + qa.z * ka.z + qa.w * ka.w;
    }
    acc *= 0.125f;  // 1/sqrt(64)
    attn[t] = acc;
    atomicMaxFloat(&amax[dst * NH + h], acc);
}

// ---------------- exp(alpha - max) + segment sum ----------------
__global__ __launch_bounds__(256)
void edge_exp_kernel(const int* __restrict__ ei, float* __restrict__ attn,
                     const float* __restrict__ amax, float* __restrict__ asum) {
    const int t = blockIdx.x * blockDim.x + threadIdx.x;
    if (t >= NE * NH) return;
    const int e = t >> 1;
    const int h = t & 1;
    const int dst = ei[NE + e];
    const float a = expf(attn[t] - amax[dst * NH + h]);
    attn[t] = a;
    atomicAdd(&asum[dst * NH + h], a);
}

// ---------------- weighted scatter of v : one wave per edge, 4 floats/lane ----------------
__global__ __launch_bounds__(256)
void edge_scatter_kernel(const int* __restrict__ ei, const float* __restrict__ attn,
                         const float* __restrict__ asum, const float* __restrict__ v,
                         float* __restrict__ agg) {
    const long long gt = (long long)blockIdx.x * blockDim.x + threadIdx.x;
    const int e = (int)(gt >> 5);
    if (e >= NE) return;
    const int lane = (int)(gt & 31);
    const int d0 = lane * 4;          // 0..124
    const int h  = d0 >> 6;           // head
    const int src = ei[e];
    const int dst = ei[NE + e];
    const float w = attn[e * NH + h] / (asum[dst * NH + h] + 1e-16f);
    const float4 vv = *(const float4*)(v + (size_t)src * HDV + d0);
    float* o = agg + (size_t)dst * HDV + d0;
    atomicAdd(o + 0, w * vv.x);
    atomicAdd(o + 1, w * vv.y);
    atomicAdd(o + 2, w * vv.z);
    atomicAdd(o + 3, w * vv.w);
}

// ---------------- (agg + skip) @ Wc + bc via WMMA, fused sum/sumsq reduction ----------------
// grid (3125), block 128 (4 waves); wave w -> column tile w (64 output cols total).
__global__ __launch_bounds__(128)
void combine_kernel(const float* __restrict__ agg, const float* __restrict__ skip,
                    const float* __restrict__ Wc, const float* __restrict__ bc,
                    float* __restrict__ y, float* __restrict__ sums) {
    const int lane = threadIdx.x & 31;
    const int wave = threadIdx.x >> 5;
    const int rb   = blockIdx.x * 16;
    const int m    = lane & 15;
    const int kh   = lane >> 4;
    const int col  = wave * 16 + m;

    const float* arow = agg  + (size_t)(rb + m) * HDV;
    const float* srow = skip + (size_t)(rb + m) * HDV;

    v8f acc = {};
#pragma unroll
    for (int kk = 0; kk < 32; ++kk) {
        const int kb = kk * 4 + kh * 2;
        v2f a, b;
        a[0] = arow[kb]     + srow[kb];
        a[1] = arow[kb + 1] + srow[kb + 1];
        b[0] = Wc[(size_t)kb * DIMV + col];
        b[1] = Wc[(size_t)(kb + 1) * DIMV + col];
        acc = __builtin_amdgcn_wmma_f32_16x16x4_f32(false, a, false, b,
                                                    (short)0, acc, false, false);
    }
    const float bias = bc[col];
    float ls = 0.0f, ls2 = 0.0f;
#pragma unroll
    for (int r = 0; r < 8; ++r) {
        const int row = rb + r + kh * 8;
        const float val = acc[r] + bias;
        y[(size_t)row * DIMV + col] = val;
        ls  += val;
        ls2 += val * val;
    }
    // wave32 reduction
#pragma unroll
    for (int off = 16; off > 0; off >>= 1) {
        ls  += __shfl_xor(ls,  off, 32);
        ls2 += __shfl_xor(ls2, off, 32);
    }
    if (lane == 0) {
        atomicAdd(&sums[0], ls);
        atomicAdd(&sums[1], ls2);
    }
}

// ---------------- graph-mode LayerNorm finalize ----------------
__global__ __launch_bounds__(256)
void finalize_kernel(const float* __restrict__ y, const float* __restrict__ sums,
                     const float* __restrict__ gamma, const float* __restrict__ beta,
                     float* __restrict__ out) {
    const int i = blockIdx.x * blockDim.x + threadIdx.x;
    if (i >= NN * DIMV) return;
    const float cnt  = (float)(NN * DIMV);
    const float mean = sums[0] / cnt;
    const float var  = fmaxf(sums[1] / cnt - mean * mean, 0.0f);
    const float inv  = 1.0f / (sqrtf(var) + 1e-5f);
    const int c = i & (DIMV - 1);
    out[i] = (y[i] - mean) * inv * gamma[c] + beta[c];
}

extern "C" void kernel_launch(void* const* d_in, const int* in_sizes, int n_in,
                              void* d_out, int out_size, void* d_ws, size_t ws_size,
                              hipStream_t stream) {
    (void)in_sizes; (void)n_in; (void)out_size; (void)ws_size;
    const float* x     = (const float*)d_in[0];
    const int*   ei    = (const int*)d_in[1];
    const float* Wq    = (const float*)d_in[2];
    const float* bq    = (const float*)d_in[3];
    const float* Wk    = (const float*)d_in[4];
    const float* bk    = (const float*)d_in[5];
    const float* Wv    = (const float*)d_in[6];
    const float* bv    = (const float*)d_in[7];
    const float* Wskip = (const float*)d_in[8];
    const float* bskip = (const float*)d_in[9];
    const float* Wc    = (const float*)d_in[10];
    const float* bc    = (const float*)d_in[11];
    const float* gamma = (const float*)d_in[12];
    const float* beta  = (const float*)d_in[13];
    float* out = (float*)d_out;

    float* ws   = (float*)d_ws;
    float* q    = ws + OFF_Q;
    float* k    = ws + OFF_K;
    float* v    = ws + OFF_V;
    float* skip = ws + OFF_SKIP;
    float* agg  = ws + OFF_AGG;
    float* attn = ws + OFF_ATTN;
    float* amax = ws + OFF_AMAX;
    float* asum = ws + OFF_ASUM;
    float* y    = ws + OFF_Y;
    float* sums = ws + OFF_SUMS;

    // 1. init accumulators
    init_kernel<<<(NN * HDV + 255) / 256, 256, 0, stream>>>(agg, amax, asum, sums);

    // 2. fused projections (WMMA)
    proj_kernel<<<dim3(NN / 16, 8), 128, 0, stream>>>(
        x, Wq, bq, Wk, bk, Wv, bv, Wskip, bskip, q, k, v, skip);

    // 3. edge logits + segment max
    edge_logits_kernel<<<(NE * NH + 255) / 256, 256, 0, stream>>>(ei, q, k, attn, amax);

    // 4. exp + segment sum
    edge_exp_kernel<<<(NE * NH + 255) / 256, 256, 0, stream>>>(ei, attn, amax, asum);

    // 5. weighted scatter-aggregate of values (one wave per edge)
    edge_scatter_kernel<<<(int)(((long long)NE * 32 + 255) / 256), 256, 0, stream>>>(
        ei, attn, asum, v, agg);

    // 6. combine GEMM (WMMA) + global moment reduction
    combine_kernel<<<NN / 16, 128, 0, stream>>>(agg, skip, Wc, bc, y, sums);

    // 7. graph LayerNorm finalize
    finalize_kernel<<<(NN * DIMV + 255) / 256, 256, 0, stream>>>(y, sums, gamma, beta, out);
}